// Hybrid_Conv2d_v2_85813446574827
// MI455X (gfx1250) — compile-verified
//
#include <hip/hip_runtime.h>

typedef __attribute__((ext_vector_type(2))) float v2f;
typedef __attribute__((ext_vector_type(4))) float v4f;
typedef __attribute__((ext_vector_type(8))) float v8f;

#define CIN   8
#define COUT  8
#define HDIM  256
#define WDIM  256
#define NCOV  3
#define KTOT  72      // CIN * 3 * 3
#define KSTEPS 18     // KTOT / 4
#define TROWS 8       // tile rows per block (1 per wave)
#define TCOLS 64      // tile cols per block (4 wmma tiles per wave)
#define NTILES 4      // TCOLS / 16
#define PROWS 10      // TROWS + 2 halo
#define PCOLS 66      // TCOLS + 2 halo
#define PSTR  96      // LDS row stride: PROWS*PSTR = 960 dwords = 15*64 -> read2st64 pairs
#define CIDELTA (PROWS * PSTR)   // dword distance between ci and ci+1 planes

// K-index bijection: kappa = 2*u + v, with u -> (c = u/9, rs = u%9), ci = 2c+v.
// A lane's two per-step values (v=0,1) are exactly CIDELTA dwords apart in LDS.

__global__ __launch_bounds__(256)
void condconv_wmma_f32(const float* __restrict__ x,
                       const float* __restrict__ cov,
                       const float* __restrict__ W0,
                       const float* __restrict__ Wj,
                       float* __restrict__ out)
{
    __shared__ float kerB[KSTEPS][32][2];          // B fragments, 4.6 KB
    __shared__ float patch[CIN * PROWS * PSTR];    // input tile, 30.7 KB

    const int t  = threadIdx.x;
    const int b  = blockIdx.z;
    const int h0 = blockIdx.y * TROWS;
    const int w0 = blockIdx.x * TCOLS;

    // ---- stage 1: per-sample dynamic kernel -> LDS as WMMA B fragments ----
    const float c0 = cov[b * NCOV + 0];
    const float c1 = cov[b * NCOV + 1];
    const float c2 = cov[b * NCOV + 2];
    for (int idx = t; idx < KSTEPS * 32 * 2; idx += 256) {
        int v    = idx & 1;
        int lane = (idx >> 1) & 31;
        int kk   = idx >> 6;
        int n    = lane & 15;                      // B column (Cout, padded)
        // kappa = 4*kk + 2*(lane>>4) + v  ->  u = 2*kk + (lane>>4), parity v
        int u    = 2 * kk + (lane >> 4);
        int c    = u / 9;
        int rs   = u - 9 * c;
        int k    = (2 * c + v) * 9 + rs;           // physical [Cin*3*3] index
        float val = 0.f;
        if (n < COUT) {
            int wi = n * KTOT + k;                 // [Cout][Cin*3*3] flat
            val = W0[wi] + c0 * Wj[0 * COUT * KTOT + wi]
                         + c1 * Wj[1 * COUT * KTOT + wi]
                         + c2 * Wj[2 * COUT * KTOT + wi];
        }
        kerB[kk][lane][v] = val;
    }

    // ---- stage 2: zero-padded input patch -> LDS ----
    const float* xb = x + (size_t)b * CIN * HDIM * WDIM;
    for (int idx = t; idx < CIN * PROWS * PCOLS; idx += 256) {
        int pc  = idx % PCOLS;
        int tmp = idx / PCOLS;
        int pr  = tmp % PROWS;
        int ci  = tmp / PROWS;
        int gh  = h0 - 1 + pr;
        int gw  = w0 - 1 + pc;
        float val = 0.f;
        if ((unsigned)gh < (unsigned)HDIM && (unsigned)gw < (unsigned)WDIM)
            val = xb[(ci * HDIM + gh) * WDIM + gw];
        patch[(ci * PROWS + pr) * PSTR + pc] = val;
    }
    __syncthreads();

    // ---- stage 3: per-wave implicit GEMM via V_WMMA_F32_16X16X4_F32 ----
    const int wave = t >> 5;          // wave's row within tile
    const int lane = t & 31;
    const int hi   = lane >> 4;
    const int m    = lane & 15;       // A row = pixel index / D column = cout

    // B fragments: loaded once, reused across all 4 16-px tiles
    v2f bfrag[KSTEPS];
#pragma unroll
    for (int kk = 0; kk < KSTEPS; ++kk) {
        bfrag[kk][0] = kerB[kk][lane][0];
        bfrag[kk][1] = kerB[kk][lane][1];
    }

    // one LDS offset per step (v=1 element is +CIDELTA from it)
    int ko[KSTEPS];
#pragma unroll
    for (int kk = 0; kk < KSTEPS; ++kk) {
        int u  = 2 * kk + hi;
        int c  = u / 9;
        int rs = u - 9 * c;
        int r  = rs / 3;
        int s  = rs - r * 3;
        ko[kk] = (2 * c * PROWS + r) * PSTR + s;   // ci-parity plane handled by +CIDELTA
    }

    const int lane_base = wave * PSTR + m;   // per-tile +16*tt folds into DS imm
    float* outb = out + (size_t)b * COUT * HDIM * WDIM;

#pragma unroll
    for (int tt = 0; tt < NTILES; ++tt) {
        const int wloc = tt * 16;

        v8f acc = {};
#pragma unroll
        for (int kk = 0; kk < KSTEPS; ++kk) {
            const float* ap = &patch[ko[kk] + lane_base + wloc];
            v2f a;
            a[0] = ap[0];          // ci = 2c
            a[1] = ap[CIDELTA];    // ci = 2c+1 : 960 dwords -> ds_load_2addr_stride64
            acc = __builtin_amdgcn_wmma_f32_16x16x4_f32(
                /*neg_a=*/false, a, /*neg_b=*/false, bfrag[kk],
                /*c_mod=*/(short)0, acc, /*reuse_a=*/false, /*reuse_b=*/false);
        }

        // D layout: vgpr v, lanes 0-15 -> M=v, lanes 16-31 -> M=v+8; N=lane&15
        if (m < COUT) {   // lanes whose cout-column < 8 store
            const int h     = h0 + wave;
            const int wbase = w0 + wloc + 8 * hi;
            float* p = outb + ((size_t)m * HDIM + h) * WDIM + wbase;
            v4f lo = { acc[0], acc[1], acc[2], acc[3] };
            v4f hv = { acc[4], acc[5], acc[6], acc[7] };
            __builtin_nontemporal_store(lo, (v4f*)p);        // keep x resident in L2
            __builtin_nontemporal_store(hv, (v4f*)(p + 4));
        }
    }
}

extern "C" void kernel_launch(void* const* d_in, const int* in_sizes, int n_in,
                              void* d_out, int out_size, void* d_ws, size_t ws_size,
                              hipStream_t stream) {
    (void)in_sizes; (void)n_in; (void)d_ws; (void)ws_size; (void)out_size;
    const float* x   = (const float*)d_in[0];   // [64, 8, 256, 256]
    const float* cov = (const float*)d_in[1];   // [64, 3]
    const float* W0  = (const float*)d_in[2];   // [8, 8, 3, 3]
    const float* Wj  = (const float*)d_in[3];   // [3, 8, 8, 3, 3]
    float* out = (float*)d_out;                 // [64, 8, 256, 256]

    dim3 grid(WDIM / TCOLS, HDIM / TROWS, 64);  // (4, 32, 64)
    condconv_wmma_f32<<<grid, 256, 0, stream>>>(x, cov, W0, Wj, out);
}